// FConvLayer_23691039605060
// MI455X (gfx1250) — compile-verified
//
#include <hip/hip_runtime.h>
#include <cstdint>

// Fixed problem dims from the reference setup_inputs()
constexpr int NB = 64, NS = 4096, NH = 128, NK = 16;
constexpr int NH4   = NH / 4;          // float4 per token row = 32
constexpr int NTILE = 64;              // tokens per block
constexpr int NHALO = NK - 1;          // 15 halo rows (circular)
constexpr int NROWS = NTILE + NHALO;   // 79 rows staged in LDS (40448 B)
constexpr int THREADS = 256;           // 8 wave32 per block
constexpr int WTOK  = NTILE / 8;       // 8 consecutive tokens per wave
constexpr float EPS   = 1e-12f;
constexpr float SCALE = 1.0f / 64.0f;  // 1/sqrt(4096) from ortho fft pair

typedef float v4f __attribute__((ext_vector_type(4)));

#if __has_builtin(__builtin_amdgcn_global_load_async_to_lds_b128) && \
    __has_builtin(__builtin_amdgcn_s_wait_asynccnt)
#define USE_ASYNC 1
// Builtin signature (from clang diagnostic): first arg is
// '<4 x int> __device__ *' i.e. v4i in addrspace(1); LDS arg is addrspace(3).
typedef int v4i __attribute__((vector_size(16)));
typedef __attribute__((address_space(1))) v4i as1_v4i;
typedef __attribute__((address_space(3))) v4i as3_v4i;
#else
#define USE_ASYNC 0
#endif

__global__ __launch_bounds__(THREADS) void fconv_ln_kernel(
    const float* __restrict__ xg, const float* __restrict__ wg,
    const float* __restrict__ lnwg, const float* __restrict__ lnbg,
    float* __restrict__ outg)
{
  __shared__ v4f lds[NROWS * NH4];

  const int tile = blockIdx.x;
  const int b    = blockIdx.y;
  const int t0   = tile * NTILE;
  const int tid  = threadIdx.x;
  const int lane = tid & 31;
  const int wave = tid >> 5;

  const v4f* x4 = (const v4f*)xg;

  // ---- Stage token rows [t0-15, t0+NTILE) into LDS (circular in seq) ----
  for (int e = tid; e < NROWS * NH4; e += THREADS) {
    const int row = e >> 5;            // / NH4
    const int c4  = e & (NH4 - 1);
    const int tok = (t0 + row - NHALO + NS) & (NS - 1);
    const v4f* src = x4 + ((size_t)b * NS + tok) * NH4 + c4;
#if USE_ASYNC
    // CDNA5 async DMA: global -> LDS without VGPR round trip (ASYNCcnt)
    __builtin_amdgcn_global_load_async_to_lds_b128(
        (as1_v4i*)(uintptr_t)src,
        (as3_v4i*)(uint32_t)(uintptr_t)&lds[e],
        /*imm offset*/0, /*cpol*/0);
#else
    lds[e] = *src;
#endif
  }

  // Overlap parameter loads with the in-flight DMA.
  const v4f* w4 = (const v4f*)wg;
  v4f wk[NK];
#pragma unroll
  for (int k = 0; k < NK; ++k) wk[k] = w4[k * NH4 + lane];
  const v4f lw = ((const v4f*)lnwg)[lane];
  const v4f lb = ((const v4f*)lnbg)[lane];

#if USE_ASYNC
  __builtin_amdgcn_s_wait_asynccnt(0);
#endif
  __syncthreads();

  // ---- Compute: wave owns 8 consecutive tokens; 16-row sliding window
  //      kept entirely in registers => 1 ds_load_b128 per new token. ----
  const int jbase = wave * WTOK;

  v4f win[16];
#pragma unroll
  for (int p = 0; p < 16; ++p)
    win[p] = lds[(jbase + p) * NH4 + lane];   // tokens t0+jbase-15 .. t0+jbase

  v4f* out4 = (v4f*)outg;
  const v4f sc = {SCALE, SCALE, SCALE, SCALE};

#pragma unroll
  for (int i = 0; i < WTOK; ++i) {
    if (i > 0)  // bring in row for token t0+jbase+i
      win[(NHALO + i) & 15] = lds[(jbase + NHALO + i) * NH4 + lane];

    v4f acc = {0.f, 0.f, 0.f, 0.f};
#pragma unroll
    for (int k = 0; k < NK; ++k)   // y[t] = sum_k w[k]*x[t-k]
      acc = __builtin_elementwise_fma(wk[k], win[(NHALO + i - k) & 15], acc);

    const v4f xin = win[(NHALO + i) & 15];               // x[t]
    const v4f h   = __builtin_elementwise_fma(acc, sc, xin);  // y/64 + x

    // LayerNorm over H=128: all channels of this token live in this wave.
    float s1 = h.x + h.y + h.z + h.w;
    float s2 = h.x * h.x + h.y * h.y + h.z * h.z + h.w * h.w;
#pragma unroll
    for (int off = 16; off > 0; off >>= 1) {
      s1 += __shfl_xor(s1, off, 32);
      s2 += __shfl_xor(s2, off, 32);
    }
    const float u   = s1 * (1.0f / NH);
    const float var = fmaf(-u, u, s2 * (1.0f / NH));     // E[h^2] - u^2
    const float rs  = rsqrtf(var + EPS);

    v4f o;
    o.x = fmaf(lw.x * rs, h.x - u, lb.x);
    o.y = fmaf(lw.y * rs, h.y - u, lb.y);
    o.z = fmaf(lw.z * rs, h.z - u, lb.z);
    o.w = fmaf(lw.w * rs, h.w - u, lb.w);

    const int t = t0 + jbase + i;
    __builtin_nontemporal_store(o, &out4[((size_t)b * NS + t) * NH4 + lane]);
  }
}

extern "C" void kernel_launch(void* const* d_in, const int* in_sizes, int n_in,
                              void* d_out, int out_size, void* d_ws, size_t ws_size,
                              hipStream_t stream) {
  (void)in_sizes; (void)n_in; (void)out_size; (void)d_ws; (void)ws_size;
  const float* x   = (const float*)d_in[0];  // [64, 4096, 128]
  const float* w   = (const float*)d_in[1];  // [1, 16, 128]
  const float* lnw = (const float*)d_in[2];  // [128]
  const float* lnb = (const float*)d_in[3];  // [128]
  float* out = (float*)d_out;                // [64, 4096, 128]

  dim3 grid(NS / NTILE, NB);   // 64 x 64 = 4096 blocks
  fconv_ln_kernel<<<grid, THREADS, 0, stream>>>(x, w, lnw, lnb, out);
}